// CriterionSA_82978768159159
// MI455X (gfx1250) — compile-verified
//
#include <hip/hip_runtime.h>
#include <math.h>

// ---------------------------------------------------------------------------
// CriterionSA (CAM + grid-PAM + CKA) for MI455X / gfx1250, f32 WMMA 16x16x4.
// Sizes: B=16, C=256, H=W=80 (N=6400), grid 5x5 -> 25 chunks of 16x16 (Nc=256),
// Ck=C/8=32. Instances p = f*400 + t*16 + b  (f: 0=S,1=T; t: chunk; b: batch).
// All WMMA GEMMs use >=2 independent accumulator chains to cover the
// WMMA->WMMA RAW latency (ISA 7.12.1) since occupancy per WGP is low.
// ---------------------------------------------------------------------------

typedef float v2f __attribute__((ext_vector_type(2)));
typedef float v8f __attribute__((ext_vector_type(8)));

#define TAU_F 1.0f
#define INV_TAU (1.0f / TAU_F)

// D = A(16x4) * B(4x16) + C   (V_WMMA_F32_16X16X4_F32)
__device__ __forceinline__ v8f wmma4(v2f a, v2f b, v8f c) {
  return __builtin_amdgcn_wmma_f32_16x16x4_f32(false, a, false, b, (short)0, c,
                                               false, false);
}

// chunk base pointer: feat[b, :, gy*16:, gx*16:]
__device__ __forceinline__ const float* chunk_ptr(const float* feat, int b,
                                                  int gy, int gx) {
  return feat + (size_t)b * 256 * 6400 + (size_t)(gy * 16) * 80 + gx * 16;
}
// element (c, n) of a chunk, n = y*16+x local
__device__ __forceinline__ size_t xf_off(int c, int n) {
  return (size_t)c * 6400 + (size_t)(n >> 4) * 80 + (n & 15);
}

// ===========================================================================
// K1: Q/K projections for all 800 instances.
//   Q[p][k][n] = sum_c Wq[k,c] * Xf[c,n] + bq[k]   (same for K with Wk,bk)
// Two accumulator chains over even/odd K slices.
// ===========================================================================
__global__ __launch_bounds__(256) void qk_proj_kernel(
    const float* featS, const float* featT, const float* Wq, const float* bq,
    const float* Wk, const float* bk, float* Qa, float* Ka) {
  int p = blockIdx.x;
  int f = p / 400, rem = p % 400, t = rem / 16, b = rem % 16;
  const float* cp = chunk_ptr(f ? featT : featS, b, t / 5, t % 5);
  int tid = threadIdx.x, wave = tid >> 5, lane = tid & 31;
  int half = lane >> 4, l16 = lane & 15;

  for (int iter = 0; iter < 8; ++iter) {
    int job = iter * 8 + wave;        // 64 tiles: {Q,K} x 2 mtiles x 16 ntiles
    int mat = job >> 5;
    int mtile = (job >> 4) & 1;
    int ntile = job & 15;
    const float* Wm = mat ? Wk : Wq;
    const float* bm = mat ? bk : bq;
    float* Dst = mat ? Ka : Qa;
    int arow = mtile * 16 + l16;
    int ncol = ntile * 16 + l16;
    v8f acc0 = {}, acc1 = {};
    for (int k0 = 0; k0 < 256; k0 += 8) {
      int ka = k0 + 2 * half;
      int kb = k0 + 4 + 2 * half;
      v2f a0 = *(const v2f*)(Wm + (size_t)arow * 256 + ka);
      v2f a1 = *(const v2f*)(Wm + (size_t)arow * 256 + kb);
      v2f b0, b1;
      b0.x = cp[xf_off(ka, ncol)];
      b0.y = cp[xf_off(ka + 1, ncol)];
      b1.x = cp[xf_off(kb, ncol)];
      b1.y = cp[xf_off(kb + 1, ncol)];
      acc0 = wmma4(a0, b0, acc0);
      acc1 = wmma4(a1, b1, acc1);
    }
    v8f acc = acc0 + acc1;
    for (int r = 0; r < 8; ++r) {
      int m = mtile * 16 + r + 8 * half;
      Dst[((size_t)p * 32 + m) * 256 + ncol] = acc[r] + bm[m];
    }
  }
}

// ===========================================================================
// K2: fused PAM per instance. One workgroup (8 waves) per instance.
//   For each block of 16 query rows:
//     energy[m,j] = sum_k Q[k,m]*K[k,j]  -> softmax over j (LDS)
//     R[c,m]      = sum_n Xf[c,n]*attn[m,n]         (attn rows sum to 1 =>)
//     pam[c,m]    = gamma*( (Wv R)[c,m] + bv[c] ) + Xf[c,m]
// Attention never touches global memory. Each wave drives two tiles in the
// same K loop sharing the common fragment (2 chains, half fragment traffic).
// ===========================================================================
__global__ __launch_bounds__(256) void pam_kernel(
    const float* featS, const float* featT, const float* Qa, const float* Ka,
    const float* Wv, const float* bv, const float* gammapam, float* pamOut) {
  __shared__ float e_s[16][258];    // energy / attn block
  __shared__ float r_s[256][18];    // R = Xf * attn^T block
  __shared__ float red_s[16][17];   // softmax reductions

  int p = blockIdx.x;
  int f = p / 400, rem = p % 400, t = rem / 16, b = rem % 16;
  const float* cp = chunk_ptr(f ? featT : featS, b, t / 5, t % 5);
  const float* Qp = Qa + (size_t)p * 32 * 256;
  const float* Kp = Ka + (size_t)p * 32 * 256;
  float* Op = pamOut + (size_t)p * 65536;
  float gamma = gammapam[0];

  int tid = threadIdx.x, wave = tid >> 5, lane = tid & 31;
  int half = lane >> 4, l16 = lane & 15;

  for (int mb = 0; mb < 16; ++mb) {
    int m_base = mb * 16;
    // --- 1) energy block [16 x 256], K = 32; two n-tiles share A(Q) frag ---
    {
      int n0 = (wave * 2) * 16 + l16;
      int n1 = (wave * 2 + 1) * 16 + l16;
      v8f acc0 = {}, acc1 = {};
      for (int k0 = 0; k0 < 32; k0 += 4) {
        int kk = k0 + 2 * half;
        v2f a, b0, b1;
        a.x = Qp[(size_t)kk * 256 + m_base + l16];
        a.y = Qp[(size_t)(kk + 1) * 256 + m_base + l16];
        b0.x = Kp[(size_t)kk * 256 + n0];
        b0.y = Kp[(size_t)(kk + 1) * 256 + n0];
        b1.x = Kp[(size_t)kk * 256 + n1];
        b1.y = Kp[(size_t)(kk + 1) * 256 + n1];
        acc0 = wmma4(a, b0, acc0);
        acc1 = wmma4(a, b1, acc1);
      }
      for (int r = 0; r < 8; ++r) {
        e_s[r + 8 * half][n0] = acc0[r];
        e_s[r + 8 * half][n1] = acc1[r];
      }
    }
    __syncthreads();
    // --- 2) stable softmax over 256 keys, 16 threads per row ---
    {
      int row = tid >> 4, j0 = tid & 15;
      float mx = -3.4e38f;
      for (int i = 0; i < 16; ++i)
        mx = fmaxf(mx, e_s[row][j0 * 16 + i]);
      red_s[row][j0] = mx;
      __syncthreads();
      float rmx = red_s[row][0];
      for (int i = 1; i < 16; ++i) rmx = fmaxf(rmx, red_s[row][i]);
      __syncthreads();
      float s = 0.f;
      for (int i = 0; i < 16; ++i) {
        float ev = expf((e_s[row][j0 * 16 + i] - rmx) * INV_TAU);
        e_s[row][j0 * 16 + i] = ev;
        s += ev;
      }
      red_s[row][j0] = s;
      __syncthreads();
      float rs = 0.f;
      for (int i = 0; i < 16; ++i) rs += red_s[row][i];
      float inv = 1.f / rs;
      for (int i = 0; i < 16; ++i) e_s[row][j0 * 16 + i] *= inv;
    }
    __syncthreads();
    // --- 3) R[c, m] = sum_n Xf[c,n]*attn[m,n]; two c-tiles share B frag ---
    {
      int c0 = wave * 16;
      int c1 = (wave + 8) * 16;
      v8f acc0 = {}, acc1 = {};
      for (int k0 = 0; k0 < 256; k0 += 4) {
        int kk = k0 + 2 * half;
        v2f bf = *(const v2f*)&e_s[l16][kk];
        v2f a0 = *(const v2f*)(cp + xf_off(c0 + l16, kk));
        v2f a1 = *(const v2f*)(cp + xf_off(c1 + l16, kk));
        acc0 = wmma4(a0, bf, acc0);
        acc1 = wmma4(a1, bf, acc1);
      }
      for (int r = 0; r < 8; ++r) {
        r_s[c0 + r + 8 * half][l16] = acc0[r];
        r_s[c1 + r + 8 * half][l16] = acc1[r];
      }
    }
    __syncthreads();
    // --- 4) out = Wv*R + bv ; pam = gamma*out + Xf; two c-tiles share B ---
    {
      int c0 = wave * 16;
      int c1 = (wave + 8) * 16;
      v8f acc0 = {}, acc1 = {};
      for (int k0 = 0; k0 < 256; k0 += 4) {
        int kk = k0 + 2 * half;
        v2f bf;
        bf.x = r_s[kk][l16];
        bf.y = r_s[kk + 1][l16];
        v2f a0 = *(const v2f*)(Wv + (size_t)(c0 + l16) * 256 + kk);
        v2f a1 = *(const v2f*)(Wv + (size_t)(c1 + l16) * 256 + kk);
        acc0 = wmma4(a0, bf, acc0);
        acc1 = wmma4(a1, bf, acc1);
      }
      int m = m_base + l16;
      for (int r = 0; r < 8; ++r) {
        int cr0 = c0 + r + 8 * half;
        int cr1 = c1 + r + 8 * half;
        Op[(size_t)cr0 * 256 + m] =
            gamma * (acc0[r] + bv[cr0]) + cp[xf_off(cr0, m)];
        Op[(size_t)cr1 * 256 + m] =
            gamma * (acc1[r] + bv[cr1]) + cp[xf_off(cr1, m)];
      }
    }
    __syncthreads();
  }
}

// ===========================================================================
// K3: per-(feat,chunk) 16x16 gram over K=65536 pam-output dims.
// A-frag == B-frag for X*X^T. Four accumulator chains over interleaved K.
// ===========================================================================
__global__ __launch_bounds__(256) void pam_gram_kernel(const float* pamOut,
                                                       float* gram) {
  __shared__ float red[8][256];
  int id = blockIdx.x;  // f*25 + t
  int f = id / 25, t = id % 25;
  const float* base = pamOut + (size_t)(f * 400 + t * 16) * 65536;
  int tid = threadIdx.x, wave = tid >> 5, lane = tid & 31;
  int half = lane >> 4, l16 = lane & 15;

  const float* rowp = base + (size_t)l16 * 65536 + (size_t)wave * 8192 +
                      2 * half;
  v8f acc0 = {}, acc1 = {}, acc2 = {}, acc3 = {};
  for (int k0 = 0; k0 < 8192; k0 += 16) {
    v2f a0 = *(const v2f*)(rowp + k0);
    v2f a1 = *(const v2f*)(rowp + k0 + 4);
    v2f a2 = *(const v2f*)(rowp + k0 + 8);
    v2f a3 = *(const v2f*)(rowp + k0 + 12);
    acc0 = wmma4(a0, a0, acc0);
    acc1 = wmma4(a1, a1, acc1);
    acc2 = wmma4(a2, a2, acc2);
    acc3 = wmma4(a3, a3, acc3);
  }
  v8f acc = (acc0 + acc1) + (acc2 + acc3);
  for (int r = 0; r < 8; ++r)
    red[wave][(r + 8 * half) * 16 + l16] = acc[r];
  __syncthreads();
  float s = 0.f;
  for (int w = 0; w < 8; ++w) s += red[w][tid];
  gram[(size_t)id * 256 + tid] = s;
}

// ===========================================================================
// K4: centered-gram CKA loss for all 25 chunks; out[1] = mean loss_PAM.
// ===========================================================================
__global__ __launch_bounds__(256) void pam_cka_kernel(const float* gram,
                                                      float* out) {
  __shared__ float gS[256], gT[256], rmS[16], rmT[16], tm[2], cS[256], cT[256];
  int tid = threadIdx.x;
  __shared__ float accum;
  if (tid == 0) accum = 0.f;
  for (int t = 0; t < 25; ++t) {
    gS[tid] = gram[(size_t)(0 * 25 + t) * 256 + tid];
    gT[tid] = gram[(size_t)(1 * 25 + t) * 256 + tid];
    __syncthreads();
    if (tid < 16) {
      float a = 0.f, b = 0.f;
      for (int j = 0; j < 16; ++j) { a += gS[tid * 16 + j]; b += gT[tid * 16 + j]; }
      rmS[tid] = a * (1.f / 16.f);
      rmT[tid] = b * (1.f / 16.f);
    }
    __syncthreads();
    if (tid == 0) {
      float a = 0.f, b = 0.f;
      for (int j = 0; j < 16; ++j) { a += rmS[j]; b += rmT[j]; }
      tm[0] = a * (1.f / 16.f);
      tm[1] = b * (1.f / 16.f);
    }
    __syncthreads();
    int i = tid >> 4, j = tid & 15;  // gram is symmetric: colmean == rowmean
    cS[tid] = gS[tid] - rmS[i] - rmS[j] + tm[0];
    cT[tid] = gT[tid] - rmT[i] - rmT[j] + tm[1];
    __syncthreads();
    if (tid == 0) {
      float h = 0.f, a = 0.f, b = 0.f;
      for (int e = 0; e < 256; ++e) {
        h += cS[e] * cT[e]; a += cS[e] * cS[e]; b += cT[e] * cT[e];
      }
      accum += -logf(fabsf(h / (sqrtf(a) * sqrtf(b))) + 1e-8f);
    }
    __syncthreads();
  }
  if (tid == 0) out[1] = accum * (1.f / 25.f);
}

// ===========================================================================
// K5: CAM energy[c,d] = sum_n X[c,n] X[d,n]  (K=6400) per (feat,batch).
// Two n-tiles per wave in one K loop sharing the A fragment.
// ===========================================================================
__global__ __launch_bounds__(256) void cam_energy_kernel(const float* featS,
                                                         const float* featT,
                                                         float* eng) {
  int id = blockIdx.x;  // f*256 + b*16 + mtile
  int f = id / 256, rem = id % 256, b = rem / 16, mt = rem % 16;
  const float* X = (f ? featT : featS) + (size_t)b * 256 * 6400;
  float* E = eng + (size_t)(f * 16 + b) * 65536;
  int tid = threadIdx.x, wave = tid >> 5, lane = tid & 31;
  int half = lane >> 4, l16 = lane & 15;

  int n0 = (wave * 2) * 16, n1 = (wave * 2 + 1) * 16;
  const float* arow = X + (size_t)(mt * 16 + l16) * 6400 + 2 * half;
  const float* brow0 = X + (size_t)(n0 + l16) * 6400 + 2 * half;
  const float* brow1 = X + (size_t)(n1 + l16) * 6400 + 2 * half;
  v8f acc0 = {}, acc1 = {};
  for (int k0 = 0; k0 < 6400; k0 += 4) {
    v2f a = *(const v2f*)(arow + k0);
    v2f b0 = *(const v2f*)(brow0 + k0);
    v2f b1 = *(const v2f*)(brow1 + k0);
    acc0 = wmma4(a, b0, acc0);
    acc1 = wmma4(a, b1, acc1);
  }
  for (int r = 0; r < 8; ++r) {
    E[(size_t)(mt * 16 + r + 8 * half) * 256 + n0 + l16] = acc0[r];
    E[(size_t)(mt * 16 + r + 8 * half) * 256 + n1 + l16] = acc1[r];
  }
}

// ===========================================================================
// K6: CAM softmax rows, in place: attn = softmax((rowmax(e) - e)/tau).
// ===========================================================================
__global__ __launch_bounds__(256) void cam_softmax_kernel(float* eng) {
  __shared__ float red[256];
  float* er = eng + (size_t)blockIdx.x * 256;
  int tid = threadIdx.x;
  float v = er[tid];
  red[tid] = v;
  __syncthreads();
  for (int off = 128; off; off >>= 1) {
    if (tid < off) red[tid] = fmaxf(red[tid], red[tid + off]);
    __syncthreads();
  }
  float m1 = red[0];
  __syncthreads();
  float en = (m1 - v) * INV_TAU;
  red[tid] = en;
  __syncthreads();
  for (int off = 128; off; off >>= 1) {
    if (tid < off) red[tid] = fmaxf(red[tid], red[tid + off]);
    __syncthreads();
  }
  float m2 = red[0];
  __syncthreads();
  float ev = expf(en - m2);
  red[tid] = ev;
  __syncthreads();
  for (int off = 128; off; off >>= 1) {
    if (tid < off) red[tid] += red[tid + off];
    __syncthreads();
  }
  er[tid] = ev / red[0];
}

// ===========================================================================
// K7: CAM out[c,n] = sum_d attn[c,d] X[d,n]; cam = gamma*out + X.
// Two m-tiles per wave in one K loop sharing the B fragment.
// ===========================================================================
__global__ __launch_bounds__(256) void cam_out_kernel(
    const float* featS, const float* featT, const float* eng,
    const float* gammacam, float* camOut) {
  int id = blockIdx.x;  // f*6400 + b*400 + ntile
  int f = id / 6400, rem = id % 6400, b = rem / 400, nt = rem % 400;
  const float* X = (f ? featT : featS) + (size_t)b * 1638400;
  const float* A = eng + (size_t)(f * 16 + b) * 65536;
  float* O = camOut + (size_t)(f * 16 + b) * 1638400;
  float g = gammacam[0];
  int tid = threadIdx.x, wave = tid >> 5, lane = tid & 31;
  int half = lane >> 4, l16 = lane & 15;
  int nbase = nt * 16;
  int m0 = wave * 16, m1 = (wave + 8) * 16;

  v8f acc0 = {}, acc1 = {};
  for (int k0 = 0; k0 < 256; k0 += 4) {
    int kk = k0 + 2 * half;
    v2f bf;
    bf.x = X[(size_t)kk * 6400 + nbase + l16];
    bf.y = X[(size_t)(kk + 1) * 6400 + nbase + l16];
    v2f a0 = *(const v2f*)(A + (size_t)(m0 + l16) * 256 + kk);
    v2f a1 = *(const v2f*)(A + (size_t)(m1 + l16) * 256 + kk);
    acc0 = wmma4(a0, bf, acc0);
    acc1 = wmma4(a1, bf, acc1);
  }
  for (int r = 0; r < 8; ++r) {
    int c0 = m0 + r + 8 * half;
    int c1 = m1 + r + 8 * half;
    size_t o0 = (size_t)c0 * 6400 + nbase + l16;
    size_t o1 = (size_t)c1 * 6400 + nbase + l16;
    O[o0] = g * acc0[r] + X[o0];
    O[o1] = g * acc1[r] + X[o1];
  }
}

// ===========================================================================
// K8: CAM 16x16 gram, split-K partials (K=1,638,400 -> 32 splits x 8 waves).
// Four accumulator chains over interleaved K.
// ===========================================================================
__global__ __launch_bounds__(256) void cam_gram_kernel(const float* camOut,
                                                       float* part) {
  __shared__ float red[8][256];
  int id = blockIdx.x;  // f*32 + split
  int f = id >> 5, sp = id & 31;
  const float* base = camOut + (size_t)f * 16 * 1638400;
  int tid = threadIdx.x, wave = tid >> 5, lane = tid & 31;
  int half = lane >> 4, l16 = lane & 15;

  const float* rowp = base + (size_t)l16 * 1638400 +
                      ((size_t)sp * 8 + wave) * 6400 + 2 * half;
  v8f acc0 = {}, acc1 = {}, acc2 = {}, acc3 = {};
  for (int k0 = 0; k0 < 6400; k0 += 16) {
    v2f a0 = *(const v2f*)(rowp + k0);
    v2f a1 = *(const v2f*)(rowp + k0 + 4);
    v2f a2 = *(const v2f*)(rowp + k0 + 8);
    v2f a3 = *(const v2f*)(rowp + k0 + 12);
    acc0 = wmma4(a0, a0, acc0);
    acc1 = wmma4(a1, a1, acc1);
    acc2 = wmma4(a2, a2, acc2);
    acc3 = wmma4(a3, a3, acc3);
  }
  v8f acc = (acc0 + acc1) + (acc2 + acc3);
  for (int r = 0; r < 8; ++r)
    red[wave][(r + 8 * half) * 16 + l16] = acc[r];
  __syncthreads();
  float s = 0.f;
  for (int w = 0; w < 8; ++w) s += red[w][tid];
  part[(size_t)id * 256 + tid] = s;
}

// ===========================================================================
// K9: reduce CAM gram partials, centered CKA; out[0] = loss_CAM.
// ===========================================================================
__global__ __launch_bounds__(256) void cam_cka_kernel(const float* part,
                                                      float* out) {
  __shared__ float gS[256], gT[256], rmS[16], rmT[16], tm[2], cS[256], cT[256];
  int tid = threadIdx.x;
  float aS = 0.f, aT = 0.f;
  for (int s = 0; s < 32; ++s) {
    aS += part[(size_t)s * 256 + tid];
    aT += part[(size_t)(32 + s) * 256 + tid];
  }
  gS[tid] = aS;
  gT[tid] = aT;
  __syncthreads();
  if (tid < 16) {
    float a = 0.f, b = 0.f;
    for (int j = 0; j < 16; ++j) { a += gS[tid * 16 + j]; b += gT[tid * 16 + j]; }
    rmS[tid] = a * (1.f / 16.f);
    rmT[tid] = b * (1.f / 16.f);
  }
  __syncthreads();
  if (tid == 0) {
    float a = 0.f, b = 0.f;
    for (int j = 0; j < 16; ++j) { a += rmS[j]; b += rmT[j]; }
    tm[0] = a * (1.f / 16.f);
    tm[1] = b * (1.f / 16.f);
  }
  __syncthreads();
  int i = tid >> 4, j = tid & 15;
  cS[tid] = gS[tid] - rmS[i] - rmS[j] + tm[0];
  cT[tid] = gT[tid] - rmT[i] - rmT[j] + tm[1];
  __syncthreads();
  if (tid == 0) {
    float h = 0.f, a = 0.f, b = 0.f;
    for (int e = 0; e < 256; ++e) {
      h += cS[e] * cT[e]; a += cS[e] * cS[e]; b += cT[e] * cT[e];
    }
    out[0] = -logf(fabsf(h / (sqrtf(a) * sqrtf(b))) + 1e-8f);
  }
}

// ===========================================================================
// Host launcher. Workspace layout (floats):
//   [0 .. 52,428,800)          bigA : pam_out (800x256x256), later cam_out
//   [52,428,800 .. +6,553,600) Qa   : Q (800x32x256); later CAM energy/attn
//   [+6,553,600 .. +6,553,600) Ka   : K (800x32x256)
//   then gramP (2x25x256), cpart (64x256). Peak ~262.3 MB.
// ===========================================================================
extern "C" void kernel_launch(void* const* d_in, const int* in_sizes, int n_in,
                              void* d_out, int out_size, void* d_ws,
                              size_t ws_size, hipStream_t stream) {
  const float* featS = (const float*)d_in[0];
  const float* featT = (const float*)d_in[1];
  const float* Wq = (const float*)d_in[2];
  const float* bq = (const float*)d_in[3];
  const float* Wk = (const float*)d_in[4];
  const float* bk = (const float*)d_in[5];
  const float* Wv = (const float*)d_in[6];
  const float* bv = (const float*)d_in[7];
  const float* gcam = (const float*)d_in[8];
  const float* gpam = (const float*)d_in[9];
  float* out = (float*)d_out;

  float* ws = (float*)d_ws;
  float* bigA = ws;                       // 52,428,800 floats
  float* Qa = ws + 52428800;              // 6,553,600 floats
  float* Ka = Qa + 6553600;               // 6,553,600 floats
  float* eng = Qa;                        // reused after K2 (needs 2,097,152)
  float* gramP = Ka + 6553600;            // 12,800 floats
  float* cpart = gramP + 12800;           // 16,384 floats

  dim3 blk(256);
  // ---- PAM pipeline ----
  qk_proj_kernel<<<dim3(800), blk, 0, stream>>>(featS, featT, Wq, bq, Wk, bk,
                                                Qa, Ka);
  pam_kernel<<<dim3(800), blk, 0, stream>>>(featS, featT, Qa, Ka, Wv, bv, gpam,
                                            bigA);
  pam_gram_kernel<<<dim3(50), blk, 0, stream>>>(bigA, gramP);
  pam_cka_kernel<<<dim3(1), blk, 0, stream>>>(gramP, out);  // out[1]
  // ---- CAM pipeline (reuses bigA and Qa regions) ----
  cam_energy_kernel<<<dim3(512), blk, 0, stream>>>(featS, featT, eng);
  cam_softmax_kernel<<<dim3(8192), blk, 0, stream>>>(eng);
  cam_out_kernel<<<dim3(12800), blk, 0, stream>>>(featS, featT, eng, gcam,
                                                  bigA);
  cam_gram_kernel<<<dim3(64), blk, 0, stream>>>(bigA, cpart);
  cam_cka_kernel<<<dim3(1), blk, 0, stream>>>(cpart, out);  // out[0]
}